// MoEGate_28802050687486
// MI455X (gfx1250) — compile-verified
//
#include <hip/hip_runtime.h>
#include <math.h>

typedef float v2f __attribute__((ext_vector_type(2)));
typedef float v8f __attribute__((ext_vector_type(8)));

#define HDIM 2048
#define NEXP 64
#define TOPK 8

// ---------------------------------------------------------------------------
// Kernel 1: router logits GEMM via V_WMMA_F32_16X16X4_F32.
// Each wave (32 lanes) computes a 16-token x 64-expert tile.
// Block = 256 threads = 8 waves => 128 token rows per block.
// ---------------------------------------------------------------------------
__global__ __launch_bounds__(256) void moe_gate_gemm(
    const float* __restrict__ x,      // [T, HDIM]
    const float* __restrict__ w,      // [NEXP, HDIM]
    float* __restrict__ logits)       // [T, NEXP]
{
    const int lane = threadIdx.x & 31;
    const int wave = threadIdx.x >> 5;
    const int rowBase = (blockIdx.x * 8 + wave) * 16;

    const int lid  = lane & 15;   // A: row M; B: col N (within 16-tile)
    const int half = lane >> 4;   // selects K-pair {0,1} vs {2,3}

    v8f acc0 = {}, acc1 = {}, acc2 = {}, acc3 = {};

    // Per-lane base pointers (contiguous float2 per k-step).
    const float* arow = x + (size_t)(rowBase + lid) * HDIM + half * 2;
    const float* b0   = w + (size_t)( 0 + lid) * HDIM + half * 2;
    const float* b1   = w + (size_t)(16 + lid) * HDIM + half * 2;
    const float* b2   = w + (size_t)(32 + lid) * HDIM + half * 2;
    const float* b3   = w + (size_t)(48 + lid) * HDIM + half * 2;

    #pragma unroll 4
    for (int k = 0; k < HDIM; k += 4) {
        v2f a   = *(const v2f*)(arow + k);
        v2f bb0 = *(const v2f*)(b0 + k);
        v2f bb1 = *(const v2f*)(b1 + k);
        v2f bb2 = *(const v2f*)(b2 + k);
        v2f bb3 = *(const v2f*)(b3 + k);
        // 8 args: (neg_a, A, neg_b, B, c_mod, C, reuse_a, reuse_b)
        acc0 = __builtin_amdgcn_wmma_f32_16x16x4_f32(false, a, false, bb0,
                                                     (short)0, acc0, false, false);
        acc1 = __builtin_amdgcn_wmma_f32_16x16x4_f32(false, a, false, bb1,
                                                     (short)0, acc1, false, false);
        acc2 = __builtin_amdgcn_wmma_f32_16x16x4_f32(false, a, false, bb2,
                                                     (short)0, acc2, false, false);
        acc3 = __builtin_amdgcn_wmma_f32_16x16x4_f32(false, a, false, bb3,
                                                     (short)0, acc3, false, false);
    }

    // D layout: VGPR r of lane l holds row (r + 8*(l>>4)), col (16*tile + l&15).
    float* outRow = logits + (size_t)(rowBase + 8 * half) * NEXP + lid;
    #pragma unroll
    for (int r = 0; r < 8; ++r) {
        outRow[(size_t)r * NEXP +  0] = acc0[r];
        outRow[(size_t)r * NEXP + 16] = acc1[r];
        outRow[(size_t)r * NEXP + 32] = acc2[r];
        outRow[(size_t)r * NEXP + 48] = acc3[r];
    }
}

// ---------------------------------------------------------------------------
// Kernel 2: per-token top-8 (insertion, first-occurrence tie-break) + softmax.
// One thread per token; logits are L2-hot from kernel 1.
// ---------------------------------------------------------------------------
__global__ __launch_bounds__(256) void moe_topk_softmax(
    const float* __restrict__ logits,   // [T, NEXP]
    float* __restrict__ weights,        // [T, TOPK]
    int*   __restrict__ indices)        // [T, TOPK]
{
    const int t = blockIdx.x * blockDim.x + threadIdx.x;
    const float* row = logits + (size_t)t * NEXP;

    float vals[TOPK];
    int   idxs[TOPK];
    #pragma unroll
    for (int i = 0; i < TOPK; ++i) { vals[i] = -INFINITY; idxs[i] = 0; }

    for (int e = 0; e < NEXP; ++e) {
        float v = row[e];
        int   vi = e;
        #pragma unroll
        for (int i = 0; i < TOPK; ++i) {
            // strict > : earlier index wins ties (matches jax.lax.top_k)
            if (v > vals[i]) {
                float tv = vals[i]; int ti = idxs[i];
                vals[i] = v;  idxs[i] = vi;
                v = tv;       vi = ti;
            }
        }
    }

    const float mx = vals[0];
    float ex[TOPK];
    float sum = 0.f;
    #pragma unroll
    for (int i = 0; i < TOPK; ++i) { ex[i] = __expf(vals[i] - mx); sum += ex[i]; }
    const float inv = 1.0f / sum;

    #pragma unroll
    for (int i = 0; i < TOPK; ++i) {
        weights[(size_t)t * TOPK + i] = ex[i] * inv;
        indices[(size_t)t * TOPK + i] = idxs[i];
    }
}

extern "C" void kernel_launch(void* const* d_in, const int* in_sizes, int n_in,
                              void* d_out, int out_size, void* d_ws, size_t ws_size,
                              hipStream_t stream) {
    const float* x = (const float*)d_in[0];   // hidden_states [4,4096,2048]
    const float* w = (const float*)d_in[1];   // gate_weight   [64,2048]
    const int T = in_sizes[0] / HDIM;         // 16384 tokens

    float* expert_w = (float*)d_out;                                  // [T,8] f32
    int*   topk_idx = (int*)((float*)d_out + (size_t)T * TOPK);       // [T,8] i32 bits
    float* logits   = (float*)d_out + (size_t)T * TOPK * 2;           // [T,64] f32

    moe_gate_gemm<<<T / 128, 256, 0, stream>>>(x, w, logits);
    moe_topk_softmax<<<T / 256, 256, 0, stream>>>(logits, expert_w, topk_idx);
}